// VectorQuantizer_42150809043547
// MI455X (gfx1250) — compile-verified
//
#include <hip/hip_runtime.h>

// Problem constants (from reference setup_inputs)
#define B_ 64
#define D_ 128
#define A_ 8
#define T_ 512
#define K_ 512
#define N_ (B_ * T_)            // 32768 tokens per agent
#define TPW 64                  // tokens per workgroup (divides T_)
#define KC 32                   // codebook rows staged in LDS per chunk
#define PITCH (D_ + 2)          // LDS row pitch in floats (pad vs bank conflicts)

// Output layout (flat, in reference return order):
// [0] q_loss, [1] e_loss, [2 .. 2+Q) quantized, [2+Q] perplexity, [3+Q ..) enc_idx
#define QELEMS 33554432         // B*D*A*T
#define OUT_PPL (2 + QELEMS)
#define OUT_ENC (3 + QELEMS)

typedef __attribute__((ext_vector_type(2))) float v2f;
typedef __attribute__((ext_vector_type(8))) float v8f;

// ws layout: ws[0] = global SSE accumulator; ws[1 .. 1+A*K) = usage counts
__global__ __launch_bounds__(256)
void vq_init(float* __restrict__ ws, float* __restrict__ out) {
  int i = blockIdx.x * blockDim.x + threadIdx.x;
  if (i < 1 + A_ * K_) ws[i] = 0.0f;
  if (i == 0) out[OUT_PPL] = 0.0f;
}

__global__ __launch_bounds__(128)
void vq_main(const float* __restrict__ x, const float* __restrict__ emb,
             float* __restrict__ out, float* __restrict__ ws) {
  __shared__ float xs[TPW][PITCH];   // token-major x tile
  __shared__ float es[KC][PITCH];    // code-major emb chunk
  __shared__ float en[KC];           // ||e||^2 per chunk code
  __shared__ float xnorm[TPW];       // ||x||^2 per token
  __shared__ int   bidx[TPW];        // best code per token
  __shared__ float bdst[TPW];        // best (||e||^2 - 2 x.e) per token

  const int tid = threadIdx.x;
  const int wg  = blockIdx.x;
  const int tilesPerAgent = N_ / TPW;          // 512
  const int a      = wg / tilesPerAgent;
  const int tile   = wg % tilesPerAgent;
  const int token0 = tile * TPW;               // global token base (= b*T + t0)
  const int b      = token0 / T_;
  const int t0     = token0 % T_;

  // ---- Stage x tile: x[b, d, a, t] -> xs[t][d]; float4 along contiguous t ----
  for (int i = 0; i < 16; ++i) {
    int flat = i * 128 + tid;                  // 0..2047
    int d  = flat >> 4;
    int t4 = flat & 15;
    const float4 v = *reinterpret_cast<const float4*>(
        x + (((size_t)b * D_ + d) * A_ + a) * T_ + t0 + t4 * 4);
    xs[t4 * 4 + 0][d] = v.x;
    xs[t4 * 4 + 1][d] = v.y;
    xs[t4 * 4 + 2][d] = v.z;
    xs[t4 * 4 + 3][d] = v.w;
  }
  __syncthreads();
  if (tid < TPW) {
    float s = 0.0f;
    for (int d = 0; d < D_; ++d) { float v = xs[tid][d]; s += v * v; }
    xnorm[tid] = s;
  }

  const int wv   = tid >> 5;                   // wave 0..3 (wave32)
  const int lane = tid & 31;
  const int l16  = lane & 15;
  const int hi   = lane >> 4;                  // 0 / 1 half of the wave
  const int koff = hi * 2;                     // f32 WMMA K pairs {0,1}/{2,3}

  float bestd[8];
  int   besti[8];
  #pragma unroll
  for (int r = 0; r < 8; ++r) { bestd[r] = 3.4e38f; besti[r] = 0; }

  for (int kc0 = 0; kc0 < K_; kc0 += KC) {
    __syncthreads();                           // prior chunk fully consumed
    // Stage emb chunk: emb[a, k, d] contiguous in d -> es[k][d]
    for (int i = 0; i < 8; ++i) {
      int flat = i * 128 + tid;                // 0..1023
      int kk = flat >> 5;
      int d4 = flat & 31;
      const float4 v = *reinterpret_cast<const float4*>(
          emb + ((size_t)a * K_ + kc0 + kk) * D_ + d4 * 4);
      *reinterpret_cast<float4*>(&es[kk][d4 * 4]) = v;
    }
    // Hint-prefetch next chunk while we compute on this one
    if (kc0 + KC < K_ && tid < KC)
      __builtin_prefetch(emb + ((size_t)a * K_ + kc0 + KC + tid) * D_, 0, 1);
    __syncthreads();
    if (tid < KC) {
      float s = 0.0f;
      for (int d = 0; d < D_; ++d) { float v = es[tid][d]; s += v * v; }
      en[tid] = s;
    }
    __syncthreads();

    #pragma unroll
    for (int ksub = 0; ksub < 2; ++ksub) {
      v8f c = {};
      const float* xrow = &xs[wv * 16 + l16][0];
      const float* erow = &es[ksub * 16 + l16][0];
      #pragma unroll
      for (int dd = 0; dd < D_; dd += 4) {
        v2f av = *reinterpret_cast<const v2f*>(xrow + dd + koff);
        v2f bv = *reinterpret_cast<const v2f*>(erow + dd + koff);
        // D = A(16x4) * B(4x16) + C, fp32 end-to-end
        c = __builtin_amdgcn_wmma_f32_16x16x4_f32(false, av, false, bv,
                                                  (short)0, c, false, false);
      }
      const int   code = kc0 + ksub * 16 + l16;
      const float ens  = en[ksub * 16 + l16];
      #pragma unroll
      for (int r = 0; r < 8; ++r) {
        float dist = ens - 2.0f * c[r];
        if (dist < bestd[r]) { bestd[r] = dist; besti[r] = code; }
      }
    }
  }

  // ---- argmin across the 16 lanes of each half-wave (C layout: N = lane&15) ----
  #pragma unroll
  for (int r = 0; r < 8; ++r) {
    float d = bestd[r]; int idx = besti[r];
    #pragma unroll
    for (int m = 1; m < 16; m <<= 1) {
      float od = __shfl_xor(d, m, 32);
      int   oi = __shfl_xor(idx, m, 32);
      if (od < d || (od == d && oi < idx)) { d = od; idx = oi; }
    }
    if (l16 == 0) {
      int tl = wv * 16 + hi * 8 + r;           // local token for this C slot
      bidx[tl] = idx;
      bdst[tl] = d;
    }
  }
  __syncthreads();

  // ---- per-token side outputs: enc indices, usage counts, SSE contribution ----
  if (tid < TPW) {
    int idx = bidx[tid];
    int n   = token0 + tid;
    out[OUT_ENC + (size_t)n * A_ + a] = (float)idx;
    atomicAdd(&ws[1 + a * K_ + idx], 1.0f);
  }
  if (tid == 0) {
    float s = 0.0f;
    for (int t = 0; t < TPW; ++t) s += bdst[t] + xnorm[t];  // = sum ||x-q||^2
    atomicAdd(&ws[0], s);
  }
  __syncthreads();

  // ---- quantized (straight-through forward value = q), coalesced along t ----
  for (int i = 0; i < 64; ++i) {
    int flat = i * 128 + tid;                  // 0..8191
    int d = flat >> 6;
    int t = flat & 63;
    float q = emb[((size_t)a * K_ + bidx[t]) * D_ + d];
    out[2 + (((size_t)b * D_ + d) * A_ + a) * T_ + t0 + t] = q;
  }
}

__global__ __launch_bounds__(256)
void vq_finalize(const float* __restrict__ ws, float* __restrict__ out) {
  __shared__ float red[256];
  const int a = blockIdx.x;
  const int tid = threadIdx.x;
  const float invN = 1.0f / (float)N_;
  float s = 0.0f;
  for (int k = tid; k < K_; k += 256) {
    float p = ws[1 + a * K_ + k] * invN;
    s += p * logf(p + 1e-10f);
  }
  red[tid] = s;
  __syncthreads();
  for (int off = 128; off > 0; off >>= 1) {
    if (tid < off) red[tid] += red[tid + off];
    __syncthreads();
  }
  if (tid == 0) atomicAdd(&out[OUT_PPL], expf(-red[0]) / (float)A_);
  if (a == 0 && tid == 0) {
    float m = ws[0] / (float)QELEMS;           // mean over A*N*D elements
    out[0] = m;                                 // q_loss
    out[1] = 0.25f * m;                         // e_loss = COMMITMENT_COST * mean
  }
}

extern "C" void kernel_launch(void* const* d_in, const int* in_sizes, int n_in,
                              void* d_out, int out_size, void* d_ws, size_t ws_size,
                              hipStream_t stream) {
  const float* x   = (const float*)d_in[0];
  const float* emb = (const float*)d_in[1];
  float* out = (float*)d_out;
  float* ws  = (float*)d_ws;

  vq_init<<<(1 + A_ * K_ + 255) / 256, 256, 0, stream>>>(ws, out);
  vq_main<<<A_ * (N_ / TPW), 128, 0, stream>>>(x, emb, out, ws);
  vq_finalize<<<A_, 256, 0, stream>>>(ws, out);
}